// GAT_15470472200431
// MI455X (gfx1250) — compile-verified
//
#include <hip/hip_runtime.h>
#include <hip/hip_bf16.h>

// ---------------------------------------------------------------------------
// GAT (2 layers) for MI455X / gfx1250, fp32 end-to-end.
// Matrix work uses V_WMMA_F32_16X16X4_F32 (true fp32 WMMA, wave32).
// ---------------------------------------------------------------------------

typedef __attribute__((ext_vector_type(2))) float v2f;
typedef __attribute__((ext_vector_type(8))) float v8f;

#define BB 4
#define NNODE 256
#define DDIM 600
#define HDIM 256

// ---------------------------------------------------------------------------
// Generic batched fp32 WMMA GEMM:  C[b] = act(A[b] @ B[b] + bias)
// One wave computes one 16x16 C tile; 4 waves per block along N tiles.
// M must be a multiple of 16, K a multiple of 4. N-edge handling:
// out-of-range lanes CLAMP their B column address (in-bounds read); their
// garbage only feeds C columns >= N, which are never stored. No branches
// in the K loop.
//
// fp32 A 16x4 per-lane layout (ISA 7.12.2):
//   lanes 0-15 : VGPR0 = A[row][k0+0], VGPR1 = A[row][k0+1]
//   lanes 16-31: VGPR0 = A[row][k0+2], VGPR1 = A[row][k0+3]
// B 4x16: same K split across lane halves, N across lanes (row-striped).
// C/D 16x16: VGPR v -> row v (lanes 0-15) / row v+8 (lanes 16-31), col=lane%16.
// ---------------------------------------------------------------------------
__global__ __launch_bounds__(128) void gat_wmma_gemm_f32(
    const float* __restrict__ A, int lda, long long strideA,
    const float* __restrict__ Bm, int ldb, long long strideB,
    float* __restrict__ C, int ldc, long long strideC,
    const float* __restrict__ bias,
    int M, int N, int K, int relu_out)
{
    const int lane  = threadIdx.x & 31;
    const int wave  = threadIdx.x >> 5;
    const int mTile = blockIdx.x;
    const int nTile = blockIdx.y * 4 + wave;
    const int nTiles = (N + 15) >> 4;
    if (nTile >= nTiles) return;

    const long long bz = blockIdx.z;
    A  += bz * strideA;
    Bm += bz * strideB;
    C  += bz * strideC;

    const int half = lane >> 4;     // which K-pair this lane holds
    const int l16  = lane & 15;
    const int row  = mTile * 16 + l16;   // A row for this lane's loads
    const int col  = nTile * 16 + l16;   // B/C column for this lane
    const bool colOK = (col < N);
    const int colClamped = colOK ? col : (N - 1);   // safe in-bounds address

    // Per-lane streaming pointers; no address math inside the K loop.
    const float* aPtr = A + (long long)row * lda + 2 * half;
    const float* bPtr = Bm + (long long)(2 * half) * ldb + colClamped;
    const long long bStep = 4LL * ldb;

    v8f acc = {};
    for (int k0 = 0; k0 < K; k0 += 4) {
        // A fragment: two consecutive K values for this lane (8B aligned).
        v2f a = *(const v2f*)aPtr;
        // B fragment: B[k0+2*half][col], B[k0+2*half+1][col]
        v2f b;
        b.x = bPtr[0];
        b.y = bPtr[ldb];
        acc = __builtin_amdgcn_wmma_f32_16x16x4_f32(
                  /*neg_a=*/false, a, /*neg_b=*/false, b,
                  /*c_mod=*/(short)0, acc,
                  /*reuse_a=*/false, /*reuse_b=*/false);
        aPtr += 4;
        bPtr += bStep;
    }

    if (!colOK) return;
    const float bv = bias ? bias[col] : 0.0f;
#pragma unroll
    for (int v = 0; v < 8; ++v) {
        const int r = mTile * 16 + v + 8 * half;
        float val = acc[v] + bv;
        if (relu_out) val = fmaxf(val, 0.0f);
        C[(long long)r * ldc + col] = val;
    }
}

// ---------------------------------------------------------------------------
// Attention scores + masked softmax for one (b, i) row.
//   e[j] = leaky_relu( sum_h relu(left[i,h] + right[j,h] + ab1[h]) * aW2[h]
//                      + ab2, 0.01 )
//   e[j] = adj==0 ? -1e9 : e[j];  att = softmax_j(e) * adj
// block = 256 threads (8 waves), one thread per j. The ReLU inside the dot
// makes this irreducibly elementwise -> VALU + LDS, not WMMA.
// ---------------------------------------------------------------------------
__global__ __launch_bounds__(256) void gat_attn_softmax(
    const float* __restrict__ left,   // [B, N, H]
    const float* __restrict__ right,  // [B, N, H]
    const float* __restrict__ adj,    // [B, N, N]
    const float* __restrict__ ab1,    // [H]
    const float* __restrict__ aW2,    // [H]
    const float* __restrict__ ab2,    // [1]
    float* __restrict__ att)          // [B, N, N]
{
    __shared__ float Ls[HDIM];
    __shared__ float W2s[HDIM];
    __shared__ float red[256];

    const int i = blockIdx.x;
    const int b = blockIdx.y;
    const int j = threadIdx.x;

    Ls[j]  = left[((long long)b * NNODE + i) * HDIM + j] + ab1[j];
    W2s[j] = aW2[j];
    __syncthreads();

    // relu-dot over H for my column j (float4-streamed right row).
    const float4* rrow =
        (const float4*)(right + ((long long)b * NNODE + j) * HDIM);
    float acc = 0.0f;
#pragma unroll 4
    for (int h4 = 0; h4 < HDIM / 4; ++h4) {
        const float4 r4 = rrow[h4];
        const int h = h4 * 4;
        acc += fmaxf(Ls[h + 0] + r4.x, 0.0f) * W2s[h + 0];
        acc += fmaxf(Ls[h + 1] + r4.y, 0.0f) * W2s[h + 1];
        acc += fmaxf(Ls[h + 2] + r4.z, 0.0f) * W2s[h + 2];
        acc += fmaxf(Ls[h + 3] + r4.w, 0.0f) * W2s[h + 3];
    }
    float e = acc + ab2[0];
    e = (e > 0.0f) ? e : 0.01f * e;            // leaky_relu(0.01)

    const float a = adj[((long long)b * NNODE + i) * NNODE + j];
    const float m = (a == 0.0f) ? -1.0e9f : e;

    // block max
    red[j] = m;
    __syncthreads();
    for (int s = 128; s > 0; s >>= 1) {
        if (j < s) red[j] = fmaxf(red[j], red[j + s]);
        __syncthreads();
    }
    const float mx = red[0];
    __syncthreads();

    const float p = __expf(m - mx);
    red[j] = p;
    __syncthreads();
    for (int s = 128; s > 0; s >>= 1) {
        if (j < s) red[j] += red[j + s];
        __syncthreads();
    }
    const float sum = red[0];

    att[((long long)b * NNODE + i) * NNODE + j] = (p / sum) * a;
}

// ---------------------------------------------------------------------------
// Host-side orchestration
// ---------------------------------------------------------------------------
static void run_layer(const float* X,           // [B,N,D] input (flattened M=1024)
                      const float* W,           // [D,D]
                      const float* bvec,        // [D]
                      const float* aW1,         // [2D,H]
                      const float* ab1,         // [H]
                      const float* aW2,         // [H]
                      const float* ab2,         // [1]
                      const float* adj,         // [B,N,N]
                      float* h_ws,              // [B,N,D]
                      float* left_ws,           // [B,N,H]
                      float* right_ws,          // [B,N,H]
                      float* att_ws,            // [B,N,N]
                      float* out,               // [B,N,D]
                      hipStream_t stream)
{
    const int M   = BB * NNODE;                  // 1024 rows
    const int nTD = (DDIM + 15) / 16;            // 38
    const int nTH = HDIM / 16;                   // 16

    // h = X @ W + b          (M=1024, N=600, K=600)
    gat_wmma_gemm_f32<<<dim3(M / 16, (nTD + 3) / 4, 1), 128, 0, stream>>>(
        X, DDIM, 0, W, DDIM, 0, h_ws, DDIM, 0, bvec, M, DDIM, DDIM, 0);

    // left  = h @ aW1[:D]    (M=1024, N=256, K=600)
    gat_wmma_gemm_f32<<<dim3(M / 16, (nTH + 3) / 4, 1), 128, 0, stream>>>(
        h_ws, DDIM, 0, aW1, HDIM, 0, left_ws, HDIM, 0, nullptr, M, HDIM, DDIM, 0);

    // right = h @ aW1[D:]    (M=1024, N=256, K=600)
    gat_wmma_gemm_f32<<<dim3(M / 16, (nTH + 3) / 4, 1), 128, 0, stream>>>(
        h_ws, DDIM, 0, aW1 + (long long)DDIM * HDIM, HDIM, 0,
        right_ws, HDIM, 0, nullptr, M, HDIM, DDIM, 0);

    // attention rows + masked softmax
    gat_attn_softmax<<<dim3(NNODE, BB), 256, 0, stream>>>(
        left_ws, right_ws, adj, ab1, aW2, ab2, att_ws);

    // out = relu(att @ h)    batched: M=256, N=600, K=256 per batch
    gat_wmma_gemm_f32<<<dim3(NNODE / 16, (nTD + 3) / 4, BB), 128, 0, stream>>>(
        att_ws, NNODE, (long long)NNODE * NNODE,
        h_ws,   DDIM,  (long long)NNODE * DDIM,
        out,    DDIM,  (long long)NNODE * DDIM,
        nullptr, NNODE, DDIM, NNODE, 1);
}

extern "C" void kernel_launch(void* const* d_in, const int* in_sizes, int n_in,
                              void* d_out, int out_size, void* d_ws, size_t ws_size,
                              hipStream_t stream)
{
    (void)in_sizes; (void)n_in; (void)out_size; (void)ws_size;

    const float* feature = (const float*)d_in[0];   // [B,N,600]
    const float* adj     = (const float*)d_in[1];   // [B,N,N]
    const float* W0      = (const float*)d_in[2];   // [600,600]
    const float* b0      = (const float*)d_in[3];   // [600]
    const float* W1      = (const float*)d_in[4];   // [600,600]
    const float* b1      = (const float*)d_in[5];   // [600]
    const float* aW1     = (const float*)d_in[6];   // [1200,256]
    const float* ab1     = (const float*)d_in[7];   // [256]
    const float* aW2     = (const float*)d_in[8];   // [256]
    const float* ab2     = (const float*)d_in[9];   // [1]

    float* out = (float*)d_out;                     // [B,N,600]

    // Workspace layout (floats). Total ~8.1 MB.
    float* ws = (float*)d_ws;
    const long long szH   = (long long)BB * NNODE * DDIM;   // 614400
    const long long szLR  = (long long)BB * NNODE * HDIM;   // 262144
    const long long szAtt = (long long)BB * NNODE * NNODE;  // 262144
    float* h_ws     = ws;
    float* left_ws  = h_ws + szH;
    float* right_ws = left_ws + szLR;
    float* att_ws   = right_ws + szLR;
    float* out1_ws  = att_ws + szAtt;                       // layer-1 output

    // Layer 1: input = feature, weights W0/b0
    run_layer(feature, W0, b0, aW1, ab1, aW2, ab2, adj,
              h_ws, left_ws, right_ws, att_ws, out1_ws, stream);

    // Layer 2: input = layer-1 output, weights W1/b1, writes d_out
    run_layer(out1_ws, W1, b1, aW1, ab1, aW2, ab2, adj,
              h_ws, left_ws, right_ws, att_ws, out, stream);
}